// SGATLayer_75488345194754
// MI455X (gfx1250) — compile-verified
//
#include <hip/hip_runtime.h>

typedef float v2f __attribute__((ext_vector_type(2)));
typedef float v8f __attribute__((ext_vector_type(8)));

#define N_NODES 50000
#define F_IN    128
#define FH      128   // F_OUT * H
#define NHEAD   8
#define FH1     136   // (F_OUT + 1) * H
#define E_EDGES 500000
#define XPAD    132   // padded LDS stride for x tile (avoid 64-bank conflicts)

// ---------------------------------------------------------------------------
// Kernel 1: fused  support0 = x@W  (WMMA f32 16x16x4),
//           attn2 = einsum(support0, a2); attn2 += sqrt(attn2^2+1)
//           attn_sup[n, f*8+h] = support0[n,f*8+h] * attn2[n,h]   (f < 16)
//           attn_sup[n, 128+h] = attn2[n,h]                       (mask col)
// Block = 256 threads (8 waves). Each block handles 16 nodes; wave w computes
// output columns [16w, 16w+16).
// ---------------------------------------------------------------------------
__global__ __launch_bounds__(256) void sgat_gemm_attn_kernel(
    const float* __restrict__ x, const float* __restrict__ Wg,
    const float* __restrict__ a2, float* __restrict__ attn_sup)
{
    __shared__ float lds_x[16 * XPAD];   // 16x128 x tile, padded
    __shared__ float sup[16 * FH1];      // staged support tile (16 x 136)
    __shared__ float attn_l[16 * NHEAD]; // per-node per-head attention

    const int tid  = threadIdx.x;
    const int wave = tid >> 5;
    const int lane = tid & 31;
    const int row0 = blockIdx.x * 16;    // N = 50000 = 3125 * 16, exact

    // ---- cooperative load of x tile (16 rows x 128 cols) into LDS ----
    for (int i = tid; i < 16 * 32; i += 256) {   // 512 float4 loads
        const int r  = i >> 5;
        const int c4 = i & 31;
        const float4 v = ((const float4*)(x + (size_t)(row0 + r) * F_IN))[c4];
        float* dst = &lds_x[r * XPAD + c4 * 4];
        dst[0] = v.x; dst[1] = v.y; dst[2] = v.z; dst[3] = v.w;
    }
    __syncthreads();

    // ---- WMMA K-loop: 32 steps of 16x16x4 (f32) ----
    const int m    = lane & 15;          // A-matrix row (M)
    const int koff = (lane >> 4) * 2;    // half-wave K pair offset
    const int col  = wave * 16 + m;      // B/D column (N)

    v8f acc = {};
    for (int k = 0; k < F_IN; k += 4) {
        v2f a, b;
        // A fragment (16x4 f32 layout): lane m, K = k+koff .. k+koff+1
        a.x = lds_x[m * XPAD + k + koff];
        a.y = lds_x[m * XPAD + k + koff + 1];
        // B fragment (4x16): rows k+koff, k+koff+1 at column `col`
        b.x = Wg[(size_t)(k + koff)     * FH + col];
        b.y = Wg[(size_t)(k + koff + 1) * FH + col];
        acc = __builtin_amdgcn_wmma_f32_16x16x4_f32(
            /*neg_a=*/false, a, /*neg_b=*/false, b,
            /*c_mod=*/(short)0, acc, /*reuse_a=*/false, /*reuse_b=*/false);
    }

    // ---- stage C tile into LDS (C layout: VGPR r -> M=r (lanes<16), M=r+8) ----
    const int rbase = (lane < 16) ? 0 : 8;
    #pragma unroll
    for (int r = 0; r < 8; ++r)
        sup[(rbase + r) * FH1 + col] = acc[r];
    __syncthreads();

    // ---- attention scalar: one thread per (node, head) ----
    if (tid < 16 * NHEAD) {
        const int node = tid >> 3;
        const int h    = tid & 7;
        float s = 0.0f;
        #pragma unroll
        for (int f = 0; f < 16; ++f)
            s += sup[node * FH1 + f * 8 + h] * a2[f * 8 + h];
        attn_l[tid] = s + sqrtf(s * s + 1.0f);
    }
    __syncthreads();

    // ---- scaled write-out of (16 x 136) tile ----
    for (int idx = tid; idx < 16 * FH1; idx += 256) {
        const int node = idx / FH1;
        const int c    = idx - node * FH1;
        const int h    = c & 7;
        const float attn = attn_l[node * NHEAD + h];
        const float v = (c < FH) ? sup[node * FH1 + c] * attn : attn;
        attn_sup[(size_t)(row0 + node) * FH1 + c] = v;
    }
}

// ---------------------------------------------------------------------------
// Kernel 2: zero the accumulator y (N x 136)
// ---------------------------------------------------------------------------
__global__ void sgat_zero_kernel(float* __restrict__ y, unsigned n)
{
    unsigned gid = blockIdx.x * blockDim.x + threadIdx.x;
    if (gid < n) y[gid] = 0.0f;
}

// ---------------------------------------------------------------------------
// Kernel 3: SpMM scatter-add.  One thread per (edge, float4 chunk): 34 chunks
// cover 136 floats.  Gather source row, scale by edge value, atomic-add into
// destination row.  The 27 MB matrices live in L2, so these are L2 atomics.
// ---------------------------------------------------------------------------
__global__ void sgat_spmm_kernel(
    const int* __restrict__ rows, const int* __restrict__ cols,
    const float* __restrict__ vals, const float* __restrict__ attn_sup,
    float* __restrict__ y)
{
    const unsigned gid = blockIdx.x * blockDim.x + threadIdx.x;
    if (gid >= (unsigned)E_EDGES * 34u) return;
    const unsigned e = gid / 34u;
    const unsigned q = gid - e * 34u;

    const int   r = rows[e];
    const int   c = cols[e];
    const float v = vals[e];

    const float4 s = *(const float4*)(attn_sup + (size_t)c * FH1 + q * 4);
    float* dst = y + (size_t)r * FH1 + q * 4;
    atomicAdd(dst + 0, v * s.x);
    atomicAdd(dst + 1, v * s.y);
    atomicAdd(dst + 2, v * s.z);
    atomicAdd(dst + 3, v * s.w);
}

// ---------------------------------------------------------------------------
// Kernel 4: normalize by propagated mask column and add bias.
// out[n, c] = y[n, c] / (y[n, 128 + (c&7)] + 1e-9) + bias[c]
// ---------------------------------------------------------------------------
__global__ void sgat_norm_kernel(
    const float* __restrict__ y, const float* __restrict__ bias,
    float* __restrict__ out)
{
    const unsigned gid = blockIdx.x * blockDim.x + threadIdx.x;
    if (gid >= (unsigned)N_NODES * FH) return;
    const unsigned n = gid >> 7;     // / 128
    const unsigned c = gid & 127;
    const float denom = y[(size_t)n * FH1 + FH + (c & 7)] + 1e-9f;
    out[gid] = y[(size_t)n * FH1 + c] / denom + bias[c];
}

// ---------------------------------------------------------------------------
extern "C" void kernel_launch(void* const* d_in, const int* in_sizes, int n_in,
                              void* d_out, int out_size, void* d_ws, size_t ws_size,
                              hipStream_t stream)
{
    (void)in_sizes; (void)n_in; (void)out_size; (void)ws_size;

    const float* x     = (const float*)d_in[0];           // (N, 128)
    const int*   eidx  = (const int*)d_in[1];             // (2, E)
    const float* vals  = (const float*)d_in[2];           // (E,)
    const float* Wg    = (const float*)d_in[3];           // (128, 128)
    const float* bias  = (const float*)d_in[4];           // (128,)
    const float* a2    = (const float*)d_in[5];           // (128,)

    const int* rows = eidx;             // edge_index[0]
    const int* cols = eidx + E_EDGES;   // edge_index[1]

    float* attn_sup = (float*)d_ws;                           // N*136 floats
    float* y        = attn_sup + (size_t)N_NODES * FH1;       // N*136 floats
    float* out      = (float*)d_out;                          // N*128 floats

    // 1) fused GEMM (WMMA f32) + attention + scale
    sgat_gemm_attn_kernel<<<N_NODES / 16, 256, 0, stream>>>(x, Wg, a2, attn_sup);

    // 2) zero accumulator (must happen every launch: graph replays reuse ws)
    {
        const unsigned n = (unsigned)N_NODES * FH1;
        sgat_zero_kernel<<<(n + 255) / 256, 256, 0, stream>>>(y, n);
    }

    // 3) SpMM scatter-add over edges
    {
        const unsigned total = (unsigned)E_EDGES * 34u;
        sgat_spmm_kernel<<<(total + 255) / 256, 256, 0, stream>>>(rows, cols, vals,
                                                                  attn_sup, y);
    }

    // 4) normalize + bias
    {
        const unsigned total = (unsigned)N_NODES * FH;
        sgat_norm_kernel<<<(total + 255) / 256, 256, 0, stream>>>(y, bias, out);
    }
}